// GCN0100_20469814133396
// MI455X (gfx1250) — compile-verified
//
#include <hip/hip_runtime.h>
#include <hip/hip_bf16.h>
#include <math.h>

// ---------------------------------------------------------------------------
// GCN 2-layer (structural + KNN graphs) forward, fp32 end-to-end.
// Dense GEMMs run on V_WMMA_F32_16X16X4_F32 (CDNA5 fp32 matrix pipe).
// ---------------------------------------------------------------------------

typedef __attribute__((ext_vector_type(2))) float v2f;
typedef __attribute__((ext_vector_type(8))) float v8f;

#define F_IN   500
#define HID    128
#define NCLS   40
#define H2PAD  48     // layer-2 GEMM N padded to 3 WMMA tiles

// ---------------------------- small utility kernels ------------------------

__global__ void fill_kernel(float* __restrict__ p, float v, int n) {
    int i = blockIdx.x * blockDim.x + threadIdx.x;
    if (i < n) p[i] = v;
}

__global__ void deg_count_kernel(const long long* __restrict__ col, long long e,
                                 float* __restrict__ deg) {
    long long i = (long long)blockIdx.x * blockDim.x + threadIdx.x;
    if (i < e) atomicAdd(&deg[(int)col[i]], 1.0f);
}

__global__ void rsqrt_kernel(float* __restrict__ p, int n) {
    int i = blockIdx.x * blockDim.x + threadIdx.x;
    if (i < n) p[i] = rsqrtf(p[i]);   // deg >= 1 always (self loop)
}

// Pack B[K, Nsrc] (row major) into Bp[(K/2), Ndst, 2] so that the (k, k+1)
// row pair at one column is a contiguous 8-byte load; zero-pads Nsrc -> Ndst.
__global__ void pack_b_kernel(const float* __restrict__ W, float* __restrict__ Wp,
                              int K, int Nsrc, int Ndst) {
    int i = blockIdx.x * blockDim.x + threadIdx.x;
    if (i >= K * Ndst) return;
    int k = i / Ndst, nn = i % Ndst;
    float v = (nn < Nsrc) ? W[k * Nsrc + nn] : 0.0f;
    Wp[(k >> 1) * (Ndst * 2) + nn * 2 + (k & 1)] = v;
}

// ------------------------------ WMMA f32 GEMM -------------------------------
// D[M, NFULL] = A[M, KDIM] @ B[KDIM, NFULL]   (B pre-packed as above)
// Each wave computes a 16-row x (NT*16)-col stripe; NFULL/(NT*16) waves share
// one 16-row tile (adjacent waves in the same block -> A tile hits WGP$).
// A layout (16x4 f32): lanes 0-15 -> VGPR0=K0,VGPR1=K1 ; lanes 16-31 -> K2,K3
// B layout (4x16 f32): VGPR0: lanes0-15 row K0, lanes16-31 row K2; VGPR1: K1/K3
// D layout: VGPR p -> row (p + 8*(lane>=16)), col = lane&15 within tile.

template <int KDIM, int NFULL, int NT>
__launch_bounds__(256, 1)
__global__ void gemm_wmma_kernel(const float* __restrict__ A,
                                 const float* __restrict__ Bp,
                                 float* __restrict__ D, int mtiles) {
    constexpr int WPT = NFULL / (NT * 16);     // waves per 16-row tile
    int wid  = blockIdx.x * (blockDim.x >> 5) + (threadIdx.x >> 5);
    int tile = wid / WPT;
    int part = wid % WPT;
    if (tile >= mtiles) return;
    int lane  = threadIdx.x & 31;
    int l15   = lane & 15;
    int khalf = lane >> 4;                     // 0: K0/K1, 1: K2/K3

    const float* ap = A + (long long)(tile * 16 + l15) * KDIM + (khalf << 1);
    const float* bp = Bp + (long long)khalf * (NFULL * 2)
                         + ((part * NT * 16) + l15) * 2;

    v8f acc[NT];
    v8f zero = {0.f, 0.f, 0.f, 0.f, 0.f, 0.f, 0.f, 0.f};
#pragma unroll
    for (int t = 0; t < NT; ++t) acc[t] = zero;

    // unroll 2: lets the scheduler overlap iteration i+1 loads with iteration
    // i WMMAs (~52-64 live VGPRs, still no spill pressure).
#pragma unroll 2
    for (int k0 = 0; k0 < KDIM; k0 += 4) {
        float2 a2 = *(const float2*)(ap + k0);            // 8B, aligned
        v2f av; av.x = a2.x; av.y = a2.y;
        const float* bk = bp + (long long)(k0 >> 1) * (NFULL * 2);
#pragma unroll
        for (int t = 0; t < NT; ++t) {
            float2 b2 = *(const float2*)(bk + t * 32);    // 8B, packed pair
            v2f bv; bv.x = b2.x; bv.y = b2.y;
            acc[t] = __builtin_amdgcn_wmma_f32_16x16x4_f32(
                false, av, false, bv, (short)0, acc[t], false, false);
        }
    }

    int mrow = tile * 16 + (khalf << 3);
    int ncol = part * NT * 16 + l15;
#pragma unroll
    for (int t = 0; t < NT; ++t) {
#pragma unroll
        for (int p = 0; p < 8; ++p)
            D[(long long)(mrow + p) * NFULL + t * 16 + ncol] = acc[t][p];
    }
}

// --------------------------- aggregation kernels ----------------------------

// R1[i, f] = H1[i, f&127] * dinv_g^2  (self-loop term for both graph halves)
__global__ void self_init1_kernel(const float* __restrict__ H1,
                                  const float* __restrict__ ds,
                                  const float* __restrict__ dk,
                                  float* __restrict__ R1, int n) {
    long long i = (long long)blockIdx.x * blockDim.x + threadIdx.x;
    long long total = (long long)n * 256;
    if (i >= total) return;
    int node = (int)(i >> 8);
    int f    = (int)(i & 255);
    float d  = (f < HID) ? ds[node] : dk[node];
    R1[i] = H1[(long long)node * HID + (f & (HID - 1))] * d * d;
}

// one wave per edge; lane-strided features; dst[col, coff+f] += h[row, f]*norm
__global__ void edge_agg_kernel(const long long* __restrict__ rows,
                                const long long* __restrict__ cols, long long e,
                                const float* __restrict__ h, int hstride,
                                const float* __restrict__ dinv,
                                float* __restrict__ dst, int dstride, int coff,
                                int F) {
    long long wid = (long long)blockIdx.x * (blockDim.x >> 5) + (threadIdx.x >> 5);
    if (wid >= e) return;
    int lane = threadIdx.x & 31;
    int r = (int)rows[wid];
    int c = (int)cols[wid];
    float norm = dinv[r] * dinv[c];
    const float* hp = h + (long long)r * hstride;
    float* dp = dst + (long long)c * dstride + coff;
    for (int f = lane; f < F; f += 32)
        atomicAdd(&dp[f], hp[f] * norm);
}

__global__ void bias_relu1_kernel(float* __restrict__ R1,
                                  const float* __restrict__ b1, int n) {
    long long i = (long long)blockIdx.x * blockDim.x + threadIdx.x;
    long long total = (long long)n * 256;
    if (i >= total) return;
    int f = (int)(i & 255);
    R1[i] = fmaxf(R1[i] + b1[f & (HID - 1)], 0.0f);
}

// R2[i, c] = H2[i, c%40] * dinv_g^2
__global__ void self_init2_kernel(const float* __restrict__ H2,
                                  const float* __restrict__ ds,
                                  const float* __restrict__ dk,
                                  float* __restrict__ R2, int n) {
    long long i = (long long)blockIdx.x * blockDim.x + threadIdx.x;
    long long total = (long long)n * (2 * NCLS);
    if (i >= total) return;
    int node = (int)(i / (2 * NCLS));
    int c    = (int)(i % (2 * NCLS));
    float d  = (c < NCLS) ? ds[node] : dk[node];
    int cc   = (c < NCLS) ? c : c - NCLS;
    R2[i] = H2[(long long)node * H2PAD + cc] * d * d;
}

__global__ void bias2_kernel(float* __restrict__ R2, const float* __restrict__ b2,
                             int n) {
    long long i = (long long)blockIdx.x * blockDim.x + threadIdx.x;
    long long total = (long long)n * (2 * NCLS);
    if (i >= total) return;
    int c = (int)(i % (2 * NCLS));
    R2[i] += b2[(c < NCLS) ? c : c - NCLS];
}

// ------------------- final linear (R2 @ Wlin^T + b) + log_softmax ------------

__global__ void final_kernel(const float* __restrict__ R2,
                             const float* __restrict__ Wlin,
                             const float* __restrict__ blin,
                             float* __restrict__ out, int n) {
    __shared__ float wl[NCLS * 2 * NCLS];   // 40 x 80
    __shared__ float bl[NCLS];
    for (int i = threadIdx.x; i < NCLS * 2 * NCLS; i += blockDim.x) wl[i] = Wlin[i];
    if (threadIdx.x < NCLS) bl[threadIdx.x] = blin[threadIdx.x];
    __syncthreads();

    int node = blockIdx.x * blockDim.x + threadIdx.x;
    if (node >= n) return;

    float r[2 * NCLS];
    const float4* rp = (const float4*)(R2 + (long long)node * (2 * NCLS));
#pragma unroll
    for (int i = 0; i < (2 * NCLS) / 4; ++i) {
        float4 v = rp[i];
        r[4 * i] = v.x; r[4 * i + 1] = v.y; r[4 * i + 2] = v.z; r[4 * i + 3] = v.w;
    }

    float logits[NCLS];
    float mx = -INFINITY;
#pragma unroll 4
    for (int c = 0; c < NCLS; ++c) {
        float s = bl[c];
        const float* w = &wl[c * 2 * NCLS];
#pragma unroll
        for (int k = 0; k < 2 * NCLS; ++k) s = fmaf(r[k], w[k], s);
        logits[c] = s;
        mx = fmaxf(mx, s);
    }
    float se = 0.0f;
#pragma unroll 4
    for (int c = 0; c < NCLS; ++c) se += __expf(logits[c] - mx);
    float lse = mx + __logf(se);
    float* op = out + (long long)node * NCLS;
#pragma unroll 4
    for (int c = 0; c < NCLS; ++c) op[c] = logits[c] - lse;
}

// --------------------------------- driver -----------------------------------

extern "C" void kernel_launch(void* const* d_in, const int* in_sizes, int n_in,
                              void* d_out, int out_size, void* d_ws, size_t ws_size,
                              hipStream_t stream) {
    const float*     x    = (const float*)d_in[0];
    const long long* ei   = (const long long*)d_in[1];   // [2, E]
    const long long* eik  = (const long long*)d_in[2];   // [2, K*N]
    const float*     W1   = (const float*)d_in[3];
    const float*     b1   = (const float*)d_in[4];
    const float*     W2   = (const float*)d_in[5];
    const float*     b2   = (const float*)d_in[6];
    const float*     Wlin = (const float*)d_in[7];
    const float*     blin = (const float*)d_in[8];

    const int n  = in_sizes[0] / F_IN;                   // 100000
    const long long E  = in_sizes[1] / 2;                // 3.2M
    const long long Ek = in_sizes[2] / 2;                // 0.5M
    const long long* e_row  = ei;
    const long long* e_col  = ei + E;
    const long long* k_row  = eik;
    const long long* k_col  = eik + Ek;

    float* ws     = (float*)d_ws;
    float* dinv_s = ws;                                  // n
    float* dinv_k = ws + n;                              // n
    float* H1     = ws + 2LL * n;                        // n*128 (reused as H2)
    float* R1     = H1 + (long long)n * HID;             // n*256
    float* R2     = R1 + (long long)n * 256;             // n*80
    float* W1p    = R2 + (long long)n * (2 * NCLS);      // 500*128 packed
    float* W2p    = W1p + F_IN * HID;                    // 256*48 packed+padded
    float* H2     = H1;                                  // reuse after R1 final

    float* out = (float*)d_out;

    const int TB  = 256;
    const int wpb = TB / 32;                             // 8 waves per block

    // --- degrees -> dinv (deg starts at 1.0 for the self loop) ---
    fill_kernel<<<(n + TB - 1) / TB, TB, 0, stream>>>(dinv_s, 1.0f, n);
    fill_kernel<<<(n + TB - 1) / TB, TB, 0, stream>>>(dinv_k, 1.0f, n);
    deg_count_kernel<<<(unsigned)((E + TB - 1) / TB), TB, 0, stream>>>(e_col, E, dinv_s);
    deg_count_kernel<<<(unsigned)((Ek + TB - 1) / TB), TB, 0, stream>>>(k_col, Ek, dinv_k);
    rsqrt_kernel<<<(n + TB - 1) / TB, TB, 0, stream>>>(dinv_s, n);
    rsqrt_kernel<<<(n + TB - 1) / TB, TB, 0, stream>>>(dinv_k, n);

    // --- pack weights for the WMMA GEMMs ---
    pack_b_kernel<<<(F_IN * HID + TB - 1) / TB, TB, 0, stream>>>(W1, W1p,
                                                                 F_IN, HID, HID);
    pack_b_kernel<<<(2 * HID * H2PAD + TB - 1) / TB, TB, 0, stream>>>(
        W2, W2p, 2 * HID, NCLS, H2PAD);

    // --- layer 1: H1 = x @ W1 (WMMA f32; 2 waves per 16-row stripe) ---
    int mtiles = n / 16;                                 // 6250
    {
        long long waves = (long long)mtiles * 2;         // WPT = 2
        gemm_wmma_kernel<F_IN, HID, 4>
            <<<(unsigned)((waves + wpb - 1) / wpb), TB, 0, stream>>>(x, W1p, H1,
                                                                     mtiles);
    }

    // --- layer 1 aggregation into R1 (struct cols 0..127, knn cols 128..255) ---
    long long tot1 = (long long)n * 256;
    self_init1_kernel<<<(unsigned)((tot1 + TB - 1) / TB), TB, 0, stream>>>(
        H1, dinv_s, dinv_k, R1, n);
    edge_agg_kernel<<<(unsigned)((E + wpb - 1) / wpb), TB, 0, stream>>>(
        e_row, e_col, E, H1, HID, dinv_s, R1, 256, 0, HID);
    edge_agg_kernel<<<(unsigned)((Ek + wpb - 1) / wpb), TB, 0, stream>>>(
        k_row, k_col, Ek, H1, HID, dinv_k, R1, 256, HID, HID);
    bias_relu1_kernel<<<(unsigned)((tot1 + TB - 1) / TB), TB, 0, stream>>>(R1, b1, n);

    // --- layer 2: H2 = R1 @ W2 (N padded 40 -> 48 = 3 WMMA tiles) ---
    gemm_wmma_kernel<2 * HID, H2PAD, 3>
        <<<(unsigned)((mtiles + wpb - 1) / wpb), TB, 0, stream>>>(R1, W2p, H2,
                                                                  mtiles);

    // --- layer 2 aggregation into R2 (struct cols 0..39, knn cols 40..79) ---
    long long tot2 = (long long)n * (2 * NCLS);
    self_init2_kernel<<<(unsigned)((tot2 + TB - 1) / TB), TB, 0, stream>>>(
        H2, dinv_s, dinv_k, R2, n);
    edge_agg_kernel<<<(unsigned)((E + wpb - 1) / wpb), TB, 0, stream>>>(
        e_row, e_col, E, H2, H2PAD, dinv_s, R2, 2 * NCLS, 0, NCLS);
    edge_agg_kernel<<<(unsigned)((Ek + wpb - 1) / wpb), TB, 0, stream>>>(
        k_row, k_col, Ek, H2, H2PAD, dinv_k, R2, 2 * NCLS, NCLS, NCLS);
    bias2_kernel<<<(unsigned)((tot2 + TB - 1) / TB), TB, 0, stream>>>(R2, b2, n);

    // --- final linear + log_softmax ---
    final_kernel<<<(n + TB - 1) / TB, TB, 0, stream>>>(R2, Wlin, blin, out, n);
}